// SSMBase_84868553769008
// MI455X (gfx1250) — compile-verified
//
#include <hip/hip_runtime.h>

// ---------------------------------------------------------------------------
// FFT long-convolution for MI455X (gfx1250), FlashFFTConv-style:
//   y[b,l,d] = irfft( rfft(x[b,:,d], 8192) * rfft(k[d,:], 8192) )[l]
//              + skip[d] * x[b,l,d]
// 8192-pt complex FFT = 3x Stockham radix-16 passes (DFT16 executed as
// V_WMMA_F32_16X16X4_F32 matmuls, full fp32) + 1 radix-2 VALU pass.
// Structural fusions:
//   * forward stage 0 reads the real zero-padded row directly: only 2 of 8
//     K-tiles are non-zero (rfft saving) -> 4 WMMAs instead of 16
//   * inverse stage 0 (m=1, twiddle==1) fuses the Y = X .* K product
//   * A-operand DFT16 fragments hoisted out of the column loop
// LDS (258 KB < 320 KB/WGP): 3 x 64KB complex buffers, 32KB twiddle table,
// 2 x 16KB staged input rows (k row reused as output staging), 2KB DFT16 tbl.
// Global<->LDS traffic via GLOBAL_LOAD_ASYNC_TO_LDS_{B32,B128} and
// GLOBAL_STORE_ASYNC_FROM_LDS_B32 (ASYNCcnt).
// One workgroup (8 wave32) per (b,d) sequence; grid = B*D = 4096 blocks.
// ---------------------------------------------------------------------------

typedef __attribute__((ext_vector_type(2))) float v2f;
typedef __attribute__((ext_vector_type(8))) float v8f;

#define LSEQ  4096   // sequence / kernel length
#define NFFT  8192   // zero-padded transform length (2L)
#define NCOL  512    // NFFT / 16 : columns per radix-16 stage
#define DCH   1024   // channels
#define BATCH 4
#define TWO_PI 6.28318530717958647692f

// ---- gfx1250 async global<->LDS copies (ASYNCcnt-tracked) ------------------
// Generic LDS pointers carry the LDS byte offset in their low 32 bits
// (flat-LDS aperture mapping), which is what the VDST/VSRC operands want.
__device__ __forceinline__ void async_load_f32(const float* gsrc, float* ldst) {
  const unsigned lds = (unsigned)(unsigned long long)ldst;
  asm volatile("global_load_async_to_lds_b32 %0, %1, off"
               :: "v"(lds), "v"(gsrc) : "memory");
}
__device__ __forceinline__ void async_load_f32x4(const float* gsrc, float* ldst) {
  const unsigned lds = (unsigned)(unsigned long long)ldst;
  asm volatile("global_load_async_to_lds_b128 %0, %1, off"
               :: "v"(lds), "v"(gsrc) : "memory");
}
__device__ __forceinline__ void async_store_f32(float* gdst, const float* lsrc) {
  const unsigned lds = (unsigned)(unsigned long long)lsrc;
  asm volatile("global_store_async_from_lds_b32 %0, %1, off"
               :: "v"(gdst), "v"(lds) : "memory");
}
__device__ __forceinline__ void wait_async0() {
  asm volatile("s_wait_asynccnt 0x0" ::: "memory");
}
__device__ __forceinline__ void wait_ds0() {
  asm volatile("s_wait_dscnt 0x0" ::: "memory");
}

// ---------------------------------------------------------------------------
// One Stockham radix-16 DIT pass over the whole NFFT-point sequence.
//   src[c + 512*q]  (c in [0,512), q in [0,16))  --DFT16 over q-->
//   dst[j + M*(r + 16*t)],  j = c mod M, t = c div M
// Input twiddle e^{fsign*i*2pi*j*q/(16M)} from the master half-table
// twtab[n] = (cos,sin)(2pi*n/8192), n = j*q*TFAC, TFAC = 8192/(16M).
// DFT16 = two real 16x16x32 matmuls chained from V_WMMA_F32_16X16X4_F32:
//   Re(D) = [Wr | -Wi] x [Xre ; Xim],   Im(D) = [Wi | Wr] x [Xre ; Xim]
// SRCMODE: 0 = complex src, 1 = real zero-padded src (only K-tiles 0..1
// non-zero; M must be 1), 2 = complex src fused with kf product (M must be 1).
// ---------------------------------------------------------------------------
template <int M, int MSHIFT, int TFAC, int SRCMODE>
__device__ __forceinline__ void radix16_stage(
    const void* __restrict__ srcv, const float2* __restrict__ kfm,
    float2* __restrict__ dst,
    const float2* __restrict__ wtab,    // DFT16 table: (cos,sin)(2pi*r*q/16)
    const float2* __restrict__ twtab,   // master twiddles, half table
    float fsign, int lane, int wave, int nwaves)
{
  const int half = lane >> 4;   // 0: lanes 0-15, 1: lanes 16-31
  const int l16  = lane & 15;
  constexpr int KTMAX = (SRCMODE == 1) ? 2 : 8;   // rfft: zero tiles skipped

  // ---- A-operand fragments: depend only on (lane, kt) -> hoisted ----
  v2f aR[KTMAX], aI[KTMAX];
#pragma unroll
  for (int kt = 0; kt < KTMAX; ++kt) {
#pragma unroll
    for (int v = 0; v < 2; ++v) {
      const int  kk = v + (half << 1) + (kt << 2);
      const int  q  = kk & 15;
      const bool hi = kk >= 16;
      const float2 w = wtab[(l16 << 4) + q];       // (cos, sin)(2pi*l16*q/16)
      const float a1 = hi ? (-fsign * w.y) : w.x;  // [Wr | -Wi]
      const float a2 = hi ? w.x : (fsign * w.y);   // [Wi |  Wr]
      if (v == 0) { aR[kt].x = a1; aI[kt].x = a2; }
      else        { aR[kt].y = a1; aI[kt].y = a2; }
    }
  }

  for (int g = wave; g < NCOL / 16; g += nwaves) {
    const int c = (g << 4) + l16;        // column this lane owns (B/D operand)
    const int j = c & (M - 1);
    const int t = c >> MSHIFT;

    v8f accR = {};   // Re of DFT16 outputs (C/D fragment)
    v8f accI = {};   // Im of DFT16 outputs

#pragma unroll
    for (int kt = 0; kt < KTMAX; ++kt) {
      v2f bf;
#pragma unroll
      for (int v = 0; v < 2; ++v) {
        // f32 4x16 B-layout: lane holds (K = v + 2*(lane/16) [+4*kt],
        // N = lane%16); packed K: rows 0..15 = Re, 16..31 = Im.
        const int  kk  = v + (half << 1) + (kt << 2);
        const int  q   = kk & 15;
        const bool hi  = kk >= 16;
        const int  idx = c + (q << 9);

        float bval;
        if (SRCMODE == 1) {
          // real zero-padded input: Im == 0 and src[idx >= LSEQ] == 0;
          // KTMAX==2 guarantees !hi and q < 8 (idx < LSEQ) here.
          bval = ((const float*)srcv)[idx];
        } else {
          float2 sv = ((const float2*)srcv)[idx];
          if (SRCMODE == 2) {            // fused Y = X .* K (m==1: no twiddle)
            const float2 kv = kfm[idx];
            sv = make_float2(sv.x * kv.x - sv.y * kv.y,
                             sv.x * kv.y + sv.y * kv.x);
          }
          if (M > 1) {                   // stage twiddle from master table
            const int    n  = j * q * TFAC;
            const float  sg = (n & 4096) ? -1.0f : 1.0f;
            const float2 tv = twtab[n & 4095];
            const float  cs = sg * tv.x;
            const float  sn = fsign * sg * tv.y;
            sv = make_float2(sv.x * cs - sv.y * sn,
                             sv.x * sn + sv.y * cs);
          }
          bval = hi ? sv.y : sv.x;
        }
        if (v == 0) bf.x = bval; else bf.y = bval;
      }
      // D = A x B + C ; full-precision fp32 matrix op, chained over K-tiles.
      accR = __builtin_amdgcn_wmma_f32_16x16x4_f32(
          false, aR[kt], false, bf, (short)0, accR, false, false);
      accI = __builtin_amdgcn_wmma_f32_16x16x4_f32(
          false, aI[kt], false, bf, (short)0, accI, false, false);
    }

    // C/D fragment layout: lane holds (M = i + 8*(lane/16), N = lane%16)
#pragma unroll
    for (int i = 0; i < 8; ++i) {
      const int r = i + (half << 3);
      dst[j + M * (r + (t << 4))] = make_float2(accR[i], accI[i]);
    }
  }
}

// Final mixed-radix factor of 2 (m = 4096): VALU butterflies, table twiddles.
__device__ __forceinline__ void radix2_stage(
    const float2* __restrict__ src, float2* __restrict__ dst,
    const float2* __restrict__ twtab, float fsign, int tid, int nthr)
{
  for (int jj = tid; jj < NFFT / 2; jj += nthr) {
    const float2 u  = src[jj];
    const float2 w  = src[jj + NFFT / 2];
    const float2 tv = twtab[jj];          // n = jj in [0, 4096)
    const float cs = tv.x;
    const float sn = fsign * tv.y;
    const float vr = w.x * cs - w.y * sn;
    const float vi = w.x * sn + w.y * cs;
    dst[jj]            = make_float2(u.x + vr, u.y + vi);
    dst[jj + NFFT / 2] = make_float2(u.x - vr, u.y - vi);
  }
}

__global__ __launch_bounds__(256, 1)
void fftconv_wmma_kernel(const float* __restrict__ x,
                         const float* __restrict__ k,
                         const float* __restrict__ skip,
                         float* __restrict__ out)
{
  extern __shared__ char smem[];
  float2* bufA   = (float2*)smem;            // 64 KB
  float2* bufB   = bufA + NFFT;              // 64 KB
  float2* kf     = bufB + NFFT;              // 64 KB (kernel spectrum)
  float*  xstage = (float*)(kf + NFFT);      // 16 KB (async-staged x row)
  float*  kstage = xstage + LSEQ;            // 16 KB (k row, later y staging)
  float2* twtab  = (float2*)(kstage + LSEQ); // 32 KB (e^{2pi i n/8192}, n<4096)
  float2* wtab   = twtab + NFFT / 2;         // 2 KB  (DFT16 (cos,sin) table)

  const int tid    = threadIdx.x;
  const int nthr   = blockDim.x;
  const int lane   = tid & 31;
  const int wave   = tid >> 5;
  const int nwaves = nthr >> 5;

  const int bd = blockIdx.x;                 // grid = B * D
  const int d  = bd & (DCH - 1);
  const int b  = bd >> 10;

  // ---- fire-and-forget global->LDS staging of both input rows ----
  const float* xrow = x + ((size_t)b * LSEQ) * DCH + d;   // stride D floats
  const float* krow = k + (size_t)d * LSEQ;               // contiguous
  for (int i = tid; i < LSEQ; i += nthr)                  // strided: B32/lane
    async_load_f32(xrow + (size_t)i * DCH, xstage + i);
  for (int i = tid; i < LSEQ / 4; i += nthr)              // contiguous: B128
    async_load_f32x4(krow + 4 * i, kstage + 4 * i);

  // ---- overlap: master twiddle table (half; tw[n+4096] = -tw[n]) ----
  for (int i = tid; i < NFFT / 2; i += nthr) {
    float sn, cs;
    __sincosf((TWO_PI / (float)NFFT) * (float)i, &sn, &cs);
    twtab[i] = make_float2(cs, sn);
  }
  // ---- DFT16 table: (cos,sin)(2*pi*r*q/16) ----
  for (int i = tid; i < 256; i += nthr) {
    const int r = i >> 4, q = i & 15;
    float sn, cs;
    __sincosf((TWO_PI / 16.0f) * (float)(r * q), &sn, &cs);
    wtab[i] = make_float2(cs, sn);
  }

  wait_async0();           // s_wait_asynccnt 0: staged rows resident
  __syncthreads();

  // ---- K = rfft(k[d,:], 8192): real stage0 (rfft saving) -> kf ----
  radix16_stage<1,   0, 512, 1>(kstage, nullptr, bufB, wtab, twtab, -1.0f, lane, wave, nwaves);
  __syncthreads();
  radix16_stage<16,  4,  32, 0>(bufB, nullptr, bufA, wtab, twtab, -1.0f, lane, wave, nwaves);
  __syncthreads();
  radix16_stage<256, 8,   2, 0>(bufA, nullptr, bufB, wtab, twtab, -1.0f, lane, wave, nwaves);
  __syncthreads();
  radix2_stage(bufB, kf, twtab, -1.0f, tid, nthr);
  __syncthreads();

  // ---- X = rfft(x[b,:,d], 8192): real stage0 -> bufA ----
  radix16_stage<1,   0, 512, 1>(xstage, nullptr, bufB, wtab, twtab, -1.0f, lane, wave, nwaves);
  __syncthreads();
  radix16_stage<16,  4,  32, 0>(bufB, nullptr, bufA, wtab, twtab, -1.0f, lane, wave, nwaves);
  __syncthreads();
  radix16_stage<256, 8,   2, 0>(bufA, nullptr, bufB, wtab, twtab, -1.0f, lane, wave, nwaves);
  __syncthreads();
  radix2_stage(bufB, bufA, twtab, -1.0f, tid, nthr);
  __syncthreads();

  // ---- inverse FFT with the Y = X .* K product fused into stage 0 ----
  radix16_stage<1,   0, 512, 2>(bufA, kf, bufB, wtab, twtab, +1.0f, lane, wave, nwaves);
  __syncthreads();
  radix16_stage<16,  4,  32, 0>(bufB, nullptr, bufA, wtab, twtab, +1.0f, lane, wave, nwaves);
  __syncthreads();
  radix16_stage<256, 8,   2, 0>(bufA, nullptr, bufB, wtab, twtab, +1.0f, lane, wave, nwaves);
  __syncthreads();
  radix2_stage(bufB, bufA, twtab, +1.0f, tid, nthr);
  __syncthreads();

  // ---- y = re(ifft)/N + skip*x into LDS, then async DMA out ----
  const float sk  = skip[d];
  const float inv = 1.0f / (float)NFFT;
  float* ystage = kstage;                     // k row no longer needed
  for (int l = tid; l < LSEQ; l += nthr)
    ystage[l] = bufA[l].x * inv + sk * xstage[l];
  wait_ds0();                                 // ystage visible to async engine

  float* orow = out + ((size_t)b * LSEQ) * DCH + d;
  for (int l = tid; l < LSEQ; l += nthr)
    async_store_f32(orow + (size_t)l * DCH, ystage + l);
  wait_async0();                              // drain before wave exit
}

// ---------------------------------------------------------------------------
// Host-side launcher. Inputs (setup_inputs order): x [B,L,D] f32,
// k [D,4096] f32, skip [D] f32. Output: [B,L,D] f32.
// ---------------------------------------------------------------------------
extern "C" void kernel_launch(void* const* d_in, const int* in_sizes, int n_in,
                              void* d_out, int out_size, void* d_ws, size_t ws_size,
                              hipStream_t stream) {
  (void)in_sizes; (void)n_in; (void)out_size; (void)d_ws; (void)ws_size;
  const float* x    = (const float*)d_in[0];
  const float* k    = (const float*)d_in[1];
  const float* skip = (const float*)d_in[2];
  float* out        = (float*)d_out;

  const dim3 grid(BATCH * DCH);   // one workgroup per (b, d) sequence
  const dim3 block(256);          // 8 wave32
  const size_t shmem = (size_t)(3 * NFFT) * sizeof(float2)   // bufA/bufB/kf
                     + (size_t)(2 * LSEQ) * sizeof(float)    // x/k(y) staging
                     + (size_t)(NFFT / 2) * sizeof(float2)   // twiddle table
                     + 256 * sizeof(float2);                 // DFT16 table

  fftconv_wmma_kernel<<<grid, block, shmem, stream>>>(x, k, skip, out);
}